// Model_10849087390519
// MI455X (gfx1250) — compile-verified
//
#include <hip/hip_runtime.h>
#include <hip/hip_bf16.h>
#include <math.h>

typedef float v2f __attribute__((ext_vector_type(2)));
typedef float v8f __attribute__((ext_vector_type(8)));

#define T_SEQ   2048
#define D_IN    1024
#define D_MODEL 1024
#define N_PROJ  3072
#define BATCH   16
#define M_TOT   (BATCH * T_SEQ)   // 32768
#define NUM_CLS 1000

// ---- GEMM1 tiling ----
#define KC    32       // K-chunk staged in LDS
#define WG_M  64       // workgroup M tile (4 M-waves)
#define WG_N  128      // workgroup N tile (2 N-waves)
#define LDA   36       // padded LDS stride (words) for A rows: 144B, 16B-aligned
#define LDB   34       // padded LDS stride (words) for B cols (transposed), even for b64
#define BUF_FLOATS (WG_M * LDA + WG_N * LDB)   // 2304 + 4352 = 6656 floats (~26KB)

// ============================================================================
// Kernel 1: proj = x @ Wp + bp, fp32 WMMA 16x16x4.
// Double-buffered LDS fed by GLOBAL_LOAD_ASYNC_TO_LDS (ASYNCcnt), so the
// global->LDS traffic of chunk k+1 overlaps the 32 WMMAs of chunk k.
// The B transpose happens inside the async copy via per-lane LDS addresses.
// ============================================================================
__global__ __launch_bounds__(256) void gemm_proj_kernel(
    const float* __restrict__ X, const float* __restrict__ Wp,
    const float* __restrict__ bp, float* __restrict__ proj)
{
    __shared__ float lds[2 * BUF_FLOATS];   // ~53 KB of 320 KB

    const int tid  = threadIdx.x;
    const int lane = tid & 31;
    const int wave = tid >> 5;
    const int mw   = wave & 3;       // 0..3  (M slot)
    const int nw   = wave >> 2;      // 0..1  (N slot)
    const int l15  = lane & 15;
    const int lh   = lane >> 4;      // lane half: K {k,k+1} vs {k+2,k+3}
    const int kb   = lh * 2;

    const int mBase = blockIdx.x * WG_M;
    const int nBase = blockIdx.y * WG_N;

    // ---- A staging plan: 64 rows x 32 K = 512 float4 -> 2 async b128 per thread
    uint32_t ldsA_addr[2], goffA[2];
    #pragma unroll
    for (int i = 0; i < 2; ++i) {
        const int s   = i * 256 + tid;     // float4 slot 0..511
        const int row = s >> 3;            // 0..63
        const int k4  = (s & 7) * 4;       // 0..28
        ldsA_addr[i] = (uint32_t)(uintptr_t)&lds[row * LDA + k4];          // 16B-aligned
        goffA[i]     = (uint32_t)(((mBase + row) * D_IN + k4) * 4);
    }
    // ---- B staging plan (transposed): 32 K x 128 N = 1024 float4 groups ->
    //      4 groups per thread, each group = 4 async b32 (one per column)
    uint32_t ldsB_addr[4], goffB[4];
    #pragma unroll
    for (int i = 0; i < 4; ++i) {
        const int s  = i * 256 + tid;      // 0..1023
        const int kk = s >> 5;             // 0..31 (constant per wave per i)
        const int c4 = (s & 31) * 4;       // 0..124
        ldsB_addr[i] = (uint32_t)(uintptr_t)&lds[WG_M * LDA + c4 * LDB + kk];
        goffB[i]     = (uint32_t)((kk * N_PROJ + nBase + c4) * 4);
    }

    auto issue_chunk = [&](int kc, int buf) {
        const uint32_t bsel = (uint32_t)(buf * (int)(BUF_FLOATS * 4));
        const uint32_t gA   = (uint32_t)(kc * 4);
        const uint32_t gB   = (uint32_t)(kc * N_PROJ * 4);
        #pragma unroll
        for (int i = 0; i < 2; ++i)
            asm volatile("global_load_async_to_lds_b128 %0, %1, %2"
                         :: "v"(ldsA_addr[i] + bsel), "v"(goffA[i] + gA), "s"(X)
                         : "memory");
        #pragma unroll
        for (int i = 0; i < 4; ++i) {
            #pragma unroll
            for (int j = 0; j < 4; ++j)
                asm volatile("global_load_async_to_lds_b32 %0, %1, %2"
                             :: "v"(ldsB_addr[i] + bsel + (uint32_t)(j * LDB * 4)),
                                "v"(goffB[i] + gB + (uint32_t)(j * 4)), "s"(Wp)
                             : "memory");
        }
    };

    v8f acc[4] = {};   // four 16x16 f32 accumulators (N tiles)

    issue_chunk(0, 0);
    asm volatile("s_wait_asynccnt 0x0" ::: "memory");
    __syncthreads();

    int cur = 0;
    for (int kc = 0; kc < D_IN; kc += KC) {
        if (kc + KC < D_IN) issue_chunk(kc + KC, cur ^ 1);   // prefetch next chunk

        const float* aRow  = &lds[cur * BUF_FLOATS + (mw * 16 + l15) * LDA];
        const float* bBase = &lds[cur * BUF_FLOATS + WG_M * LDA];
        #pragma unroll
        for (int k0 = 0; k0 < KC; k0 += 4) {
            const v2f a = *(const v2f*)(aRow + k0 + kb);
            #pragma unroll
            for (int nt = 0; nt < 4; ++nt) {
                const v2f b = *(const v2f*)(bBase + (nw * 64 + nt * 16 + l15) * LDB + k0 + kb);
                acc[nt] = __builtin_amdgcn_wmma_f32_16x16x4_f32(
                    false, a, false, b, (short)0, acc[nt], false, false);
            }
        }
        asm volatile("s_wait_asynccnt 0x0" ::: "memory");
        __syncthreads();
        cur ^= 1;
    }

    // ---- epilogue: bias add + store ----
    #pragma unroll
    for (int nt = 0; nt < 4; ++nt) {
        const int col  = nBase + nw * 64 + nt * 16 + l15;
        const float bv = bp[col];
        #pragma unroll
        for (int v = 0; v < 8; ++v) {
            const int m = mBase + mw * 16 + v + lh * 8;
            proj[(size_t)m * N_PROJ + col] = acc[nt][v] + bv;
        }
    }
}

// ============================================================================
// Kernel 2: per-(b,d) sequential scan; only last timestep's c*o is needed.
// ============================================================================
__global__ __launch_bounds__(256) void scan_kernel(
    const float* __restrict__ proj, float* __restrict__ hn)
{
    const int g = blockIdx.x * blockDim.x + threadIdx.x;   // 0..16383
    const int b = g >> 10;
    const int d = g & 1023;
    const float* base = proj + (size_t)b * T_SEQ * N_PROJ;

    float c = 0.f;
    for (int t = 0; t < T_SEQ; ++t) {
        const float fh = base[(size_t)t * N_PROJ + d];
        const float zh = base[(size_t)t * N_PROJ + D_MODEL + d];
        const float f  = 1.f / (1.f + expf(-fh));
        const float z  = (1.f - f) * tanhf(zh);
        c = f * c + z;
    }
    const float oh = base[(size_t)(T_SEQ - 1) * N_PROJ + 2 * D_MODEL + d];
    const float o  = 1.f / (1.f + expf(-oh));
    hn[g] = c * o;
}

// ============================================================================
// Kernel 3: small MLP layer: out[16,outN] = act(A[16,K] @ W[K,outN] + bias)
// One wave per 16-column output tile, fp32 WMMA over K.
// ============================================================================
__global__ __launch_bounds__(128) void mlp_kernel(
    const float* __restrict__ A, const float* __restrict__ W,
    const float* __restrict__ bias, float* __restrict__ out,
    int K, int Npad, int outN, int doRelu)
{
    const int lane = threadIdx.x & 31;
    const int wave = blockIdx.x * (blockDim.x >> 5) + (threadIdx.x >> 5);
    const int n0   = wave * 16;
    if (n0 >= Npad) return;                    // whole-wave early exit keeps EXEC uniform

    const int l15 = lane & 15;
    const int lh  = lane >> 4;
    const int kbo = lh * 2;
    const int n   = n0 + l15;
    const int nc  = (n < outN) ? n : (outN - 1);   // clamp loads for ragged last tile

    v8f acc = {};
    for (int k = 0; k < K; k += 4) {
        v2f a, b;
        a.x = A[l15 * K + k + kbo];
        a.y = A[l15 * K + k + kbo + 1];
        b.x = W[(size_t)(k + kbo) * outN + nc];
        b.y = W[(size_t)(k + kbo + 1) * outN + nc];
        acc = __builtin_amdgcn_wmma_f32_16x16x4_f32(
            false, a, false, b, (short)0, acc, false, false);
    }

    const float bv = bias[nc];
    #pragma unroll
    for (int v = 0; v < 8; ++v) {
        const int m = v + lh * 8;
        float r = acc[v] + bv;
        if (doRelu) r = fmaxf(r, 0.f);
        if (n < outN) out[(size_t)m * outN + n] = r;
    }
}

// ============================================================================
extern "C" void kernel_launch(void* const* d_in, const int* in_sizes, int n_in,
                              void* d_out, int out_size, void* d_ws, size_t ws_size,
                              hipStream_t stream) {
    const float* x  = (const float*)d_in[0];   // [16,2048,1024]
    const float* Wp = (const float*)d_in[1];   // [1024,3072]
    const float* bp = (const float*)d_in[2];   // [3072]
    const float* W0 = (const float*)d_in[3];   // [1024,2048]
    const float* b0 = (const float*)d_in[4];   // [2048]
    const float* W1 = (const float*)d_in[5];   // [2048,2048]
    const float* b1 = (const float*)d_in[6];   // [2048]
    const float* W2 = (const float*)d_in[7];   // [2048,1000]
    const float* b2 = (const float*)d_in[8];   // [1000]
    float* out = (float*)d_out;                // [16,1000]

    // workspace layout (floats)
    float* proj = (float*)d_ws;                          // 32768*3072
    float* hn   = proj + (size_t)M_TOT * N_PROJ;         // 16*1024
    float* h1   = hn + BATCH * D_MODEL;                  // 16*2048
    float* h2   = h1 + BATCH * 2048;                     // 16*2048

    // 1) projection GEMM
    dim3 g1(M_TOT / WG_M, N_PROJ / WG_N);                // 512 x 24
    gemm_proj_kernel<<<g1, 256, 0, stream>>>(x, Wp, bp, proj);

    // 2) recurrence scan -> hn[16,1024]
    scan_kernel<<<(BATCH * D_MODEL) / 256, 256, 0, stream>>>(proj, hn);

    // 3) MLP: relu(hn@W0+b0) -> relu(@W1+b1) -> @W2+b2
    mlp_kernel<<<(2048 / 16) / 4, 128, 0, stream>>>(hn, W0, b0, h1, 1024, 2048, 2048, 1);
    mlp_kernel<<<(2048 / 16) / 4, 128, 0, stream>>>(h1, W1, b1, h2, 2048, 2048, 2048, 1);
    const int npad2 = ((NUM_CLS + 15) / 16) * 16;        // 1008
    mlp_kernel<<<(npad2 / 16 + 3) / 4, 128, 0, stream>>>(h2, W2, b2, out, 2048, npad2, NUM_CLS, 0);
}